// GraphSAGENetwork_66391604461927
// MI455X (gfx1250) — compile-verified
//
#include <hip/hip_runtime.h>
#include <hip/hip_bf16.h>
#include <math.h>

// ---------------------------------------------------------------------------
// GraphSAGE (3-layer) for MI455X / gfx1250.
//   agg pass  : wave-per-edge gather + global_atomic_add_f32 scatter (L2-resident)
//   gemm pass : fused  mean@Wl^T + x@Wr^T + b  via V_WMMA_F32_16X16X4_F32 (fp32 exact)
//   epilogue  : wave-per-row log_softmax with wave32 shuffle reductions
// Out-of-range rows/cols are handled only at store time: a WMMA output element
// D[m][n] depends only on A row m and B col n, so clamped (in-bounds) loads of
// "don't care" rows/cols cannot pollute valid outputs.
// ---------------------------------------------------------------------------

typedef __attribute__((ext_vector_type(2))) float v2f;
typedef __attribute__((ext_vector_type(8))) float v8f;

#define F_K 128   // F_IN == H == 128 (GEMM reduction width, fixed)

__global__ void fill_zero_f32(float* __restrict__ p, long n) {
  long i = (long)blockIdx.x * blockDim.x + threadIdx.x;
  if (i < n) p[i] = 0.0f;
}

__global__ void count_edges_kernel(const int* __restrict__ dst,
                                   float* __restrict__ cnt, int E) {
  int e = blockIdx.x * blockDim.x + threadIdx.x;
  if (e < E) atomicAdd(&cnt[dst[e]], 1.0f);
}

__global__ void make_inv_kernel(float* __restrict__ cnt, int N) {
  int i = blockIdx.x * blockDim.x + threadIdx.x;
  if (i < N) cnt[i] = 1.0f / fmaxf(cnt[i], 1.0f);
}

// One wave (32 lanes) per edge; each lane moves 4 consecutive floats (float4).
__global__ void aggregate_kernel(const float* __restrict__ x,
                                 const int* __restrict__ src,
                                 const int* __restrict__ dst,
                                 float* __restrict__ agg, int E) {
  const int lane = threadIdx.x & 31;
  const int e = (int)(((long)blockIdx.x * blockDim.x + threadIdx.x) >> 5);
  if (e >= E) return;
  const int s = src[e];
  const int d = dst[e];
  const float4 v = ((const float4*)(x + (size_t)s * F_K))[lane];
  float* a = agg + (size_t)d * F_K + lane * 4;
  atomicAdd(a + 0, v.x);   // global_atomic_add_f32, no return -> STOREcnt
  atomicAdd(a + 1, v.y);
  atomicAdd(a + 2, v.z);
  atomicAdd(a + 3, v.w);
}

// out[N,Hout] = (agg*inv) @ Wl^T + xs @ Wr^T + bias  (+ optional ReLU)
// block = 128 threads = 4 waves; wave w owns 16x16 tile at
//   rows [blockIdx.x*16, +16), cols [(blockIdx.y*4 + w)*16, +16)
// WMMA fp32 K=4 operand layout: lane l, vgpr v -> K = v + 2*(l>>4), M/N = l&15.
__global__ void sage_gemm_wmma(const float* __restrict__ agg,
                               const float* __restrict__ inv,
                               const float* __restrict__ xs,
                               const float* __restrict__ Wl,
                               const float* __restrict__ Wr,
                               const float* __restrict__ bias,
                               float* __restrict__ out,
                               int N, int Hout, int relu) {
  const int lane = threadIdx.x & 31;
  const int wave = threadIdx.x >> 5;
  const int half = lane >> 4;     // 0 or 1
  const int mr   = lane & 15;
  const int tm   = blockIdx.x * 16;
  const int tn   = (blockIdx.y * 4 + wave) * 16;
  const int row  = tm + mr;       // A-matrix M index for this lane
  const int col  = tn + mr;       // B-matrix N index (weight row) for this lane
  const bool rowOK = row < N;
  const bool colOK = col < Hout;

  const float invc = rowOK ? inv[row] : 0.0f;
  // Clamp OOB lanes to row/col 0: loads stay in-bounds; the resulting
  // "don't care" D rows/cols are simply never stored.
  const float* __restrict__ ap = agg + (size_t)(rowOK ? row : 0) * F_K + 2 * half;
  const float* __restrict__ xp = xs  + (size_t)(rowOK ? row : 0) * F_K + 2 * half;
  const float* __restrict__ lp = Wl  + (size_t)(colOK ? col : 0) * F_K + 2 * half;
  const float* __restrict__ rp = Wr  + (size_t)(colOK ? col : 0) * F_K + 2 * half;

  v8f acc = {0.f, 0.f, 0.f, 0.f, 0.f, 0.f, 0.f, 0.f};

#pragma unroll 4
  for (int k0 = 0; k0 < F_K; k0 += 4) {
    const float2 am = *(const float2*)(ap + k0);
    const float2 ax = *(const float2*)(xp + k0);
    const float2 bl = *(const float2*)(lp + k0);
    const float2 br = *(const float2*)(rp + k0);

    v2f a1; a1[0] = am.x * invc; a1[1] = am.y * invc;   // mean = agg * 1/deg
    v2f a2; a2[0] = ax.x;        a2[1] = ax.y;
    v2f b1; b1[0] = bl.x;        b1[1] = bl.y;
    v2f b2; b2[0] = br.x;        b2[1] = br.y;

    // D = A*B + C ; fp32 exact path, two WMMAs per K-step (mean-term + self-term)
    acc = __builtin_amdgcn_wmma_f32_16x16x4_f32(false, a1, false, b1,
                                                (short)0, acc, false, false);
    acc = __builtin_amdgcn_wmma_f32_16x16x4_f32(false, a2, false, b2,
                                                (short)0, acc, false, false);
  }

  const float bv = colOK ? bias[col] : 0.0f;
#pragma unroll
  for (int v = 0; v < 8; ++v) {
    float t = acc[v] + bv;
    if (relu) t = fmaxf(t, 0.0f);
    const int r = tm + v + 8 * half;        // D: M = v + 8*(lane>>4), N = lane&15
    if (r < N && colOK) out[(size_t)r * Hout + col] = t;
  }
}

// In-place log_softmax over rows of y[N,C], one wave32 per row (C <= 64).
__global__ void log_softmax_rows(float* __restrict__ y, int N, int C) {
  const int lane = threadIdx.x & 31;
  const int wave = threadIdx.x >> 5;
  const int node = blockIdx.x * (blockDim.x >> 5) + wave;
  if (node >= N) return;
  float* row = y + (size_t)node * C;
  const float v0 = (lane < C)      ? row[lane]      : -INFINITY;
  const float v1 = (lane + 32 < C) ? row[lane + 32] : -INFINITY;
  float m = fmaxf(v0, v1);
#pragma unroll
  for (int o = 16; o > 0; o >>= 1) m = fmaxf(m, __shfl_xor(m, o, 32));
  float s = ((lane < C) ? expf(v0 - m) : 0.0f) +
            ((lane + 32 < C) ? expf(v1 - m) : 0.0f);
#pragma unroll
  for (int o = 16; o > 0; o >>= 1) s += __shfl_xor(s, o, 32);
  const float lse = m + logf(s);
  if (lane < C)      row[lane]      = v0 - lse;
  if (lane + 32 < C) row[lane + 32] = v1 - lse;
}

extern "C" void kernel_launch(void* const* d_in, const int* in_sizes, int n_in,
                              void* d_out, int out_size, void* d_ws, size_t ws_size,
                              hipStream_t stream) {
  const float* x   = (const float*)d_in[0];
  const int*   ei  = (const int*)d_in[1];   // edge_index [2,E] (int32 on device)
  const float* Wl1 = (const float*)d_in[2];
  const float* Wr1 = (const float*)d_in[3];
  const float* b1  = (const float*)d_in[4];
  const float* Wl2 = (const float*)d_in[5];
  const float* Wr2 = (const float*)d_in[6];
  const float* b2  = (const float*)d_in[7];
  const float* Wl3 = (const float*)d_in[8];
  const float* Wr3 = (const float*)d_in[9];
  const float* b3  = (const float*)d_in[10];
  float* out = (float*)d_out;

  const int N = in_sizes[0] / F_K;   // 50000
  const int E = in_sizes[1] / 2;     // 800000
  const int H = in_sizes[4];         // 128
  const int C = in_sizes[10];        // 40

  const int* src = ei;
  const int* dst = ei + E;

  // Workspace layout (floats): inv[N] | agg[N*128] | h1[N*128] | h2[N*128]
  float* inv = (float*)d_ws;
  float* agg = inv + N;
  float* h1  = agg + (size_t)N * F_K;
  float* h2  = h1  + (size_t)N * F_K;

  const long aggN = (long)N * F_K;
  const dim3 blk256(256);

  // Degrees -> 1/max(cnt,1)   (layer-invariant)
  fill_zero_f32<<<(unsigned)((N + 255) / 256), blk256, 0, stream>>>(inv, N);
  count_edges_kernel<<<(unsigned)((E + 255) / 256), blk256, 0, stream>>>(dst, inv, E);
  make_inv_kernel<<<(unsigned)((N + 255) / 256), blk256, 0, stream>>>(inv, N);

  const unsigned aggBlocks  = (unsigned)((E + 7) / 8);           // 8 edge-waves / block
  const unsigned zeroBlocks = (unsigned)((aggN + 255) / 256);
  const dim3 gemmBlk(128);                                        // 4 waves
  const dim3 gH((unsigned)((N + 15) / 16), (unsigned)((H + 63) / 64));
  const dim3 gC((unsigned)((N + 15) / 16), (unsigned)((C + 63) / 64));

  // ---- Layer 1: x -> h1 (ReLU) ----
  fill_zero_f32<<<zeroBlocks, blk256, 0, stream>>>(agg, aggN);
  aggregate_kernel<<<aggBlocks, blk256, 0, stream>>>(x, src, dst, agg, E);
  sage_gemm_wmma<<<gH, gemmBlk, 0, stream>>>(agg, inv, x, Wl1, Wr1, b1, h1, N, H, 1);

  // ---- Layer 2: h1 -> h2 (ReLU) ----
  fill_zero_f32<<<zeroBlocks, blk256, 0, stream>>>(agg, aggN);
  aggregate_kernel<<<aggBlocks, blk256, 0, stream>>>(h1, src, dst, agg, E);
  sage_gemm_wmma<<<gH, gemmBlk, 0, stream>>>(agg, inv, h1, Wl2, Wr2, b2, h2, N, H, 1);

  // ---- Layer 3: h2 -> d_out (no ReLU), then in-place log_softmax ----
  fill_zero_f32<<<zeroBlocks, blk256, 0, stream>>>(agg, aggN);
  aggregate_kernel<<<aggBlocks, blk256, 0, stream>>>(h2, src, dst, agg, E);
  sage_gemm_wmma<<<gC, gemmBlk, 0, stream>>>(agg, inv, h2, Wl3, Wr3, b3, out, N, C, 0);

  log_softmax_rows<<<(unsigned)((N + 7) / 8), blk256, 0, stream>>>(out, N, C);
}